// WindowAttention_206158430371
// MI455X (gfx1250) — compile-verified
//
#include <hip/hip_runtime.h>
#include <hip/hip_bf16.h>
#include <math.h>

// ---------------- problem constants ----------------
#define BWIN   2048
#define NTOK   49
#define DIMC   384
#define NH     12
#define HD     32
#define NWIN   64
#define TROWS  169      // (2*7-1)^2
#define CPBH   512

// LDS row strides (floats). 388 = 384+4 pad to break 64-bank conflicts on
// column-strided A-fragment reads; 36/52 similar for the small tiles.
#define XS 388
#define QS 36
#define AS 52

typedef float v2f __attribute__((ext_vector_type(2)));
typedef float v8f __attribute__((ext_vector_type(8)));

// D = A(16x4 f32) * B(4x16 f32) + C(16x16 f32)  -- full-precision CDNA5 WMMA.
static __device__ __forceinline__ v8f wmma4(v2f a, v2f b, v8f c) {
    return __builtin_amdgcn_wmma_f32_16x16x4_f32(
        /*neg_a=*/false, a, /*neg_b=*/false, b,
        /*c_mod=*/(short)0, c, /*reuse_a=*/false, /*reuse_b=*/false);
}

// ---------------------------------------------------------------------------
// Kernel 1: continuous relative-position-bias MLP.
// bias_table[t*12+h] = 16*sigmoid( relu(tbl[t]@w1.T + b1) @ w2[h] )
// 169*12 outputs, each a 512-length reduction: trivial VALU work.
// ---------------------------------------------------------------------------
__global__ void cpb_table_kernel(const float* __restrict__ tbl,
                                 const float* __restrict__ w1,
                                 const float* __restrict__ b1,
                                 const float* __restrict__ w2,
                                 float* __restrict__ bias_table) {
    int gid = blockIdx.x * blockDim.x + threadIdx.x;
    if (gid >= TROWS * NH) return;
    int t = gid / NH, h = gid % NH;
    float c0 = tbl[t * 2 + 0];
    float c1 = tbl[t * 2 + 1];
    float acc = 0.f;
    for (int k = 0; k < CPBH; ++k) {
        float hx = fmaxf(c0 * w1[k * 2 + 0] + c1 * w1[k * 2 + 1] + b1[k], 0.f);
        acc += hx * w2[h * CPBH + k];
    }
    bias_table[gid] = 16.0f / (1.0f + expf(-acc));
}

// ---------------------------------------------------------------------------
// Kernel 2: fully-fused window attention. One block per window, 384 threads
// = 12 wave32 waves. All intermediates in LDS (~274 KB); qkv/proj weights
// streamed from L2. fp32 WMMA throughout (memory-bound kernel -> keep full
// precision).
// ---------------------------------------------------------------------------
__global__ __launch_bounds__(384, 1)
void swinv2_attn_fused(const float* __restrict__ x,
                       const float* __restrict__ mask,
                       const float* __restrict__ qkv_w,
                       const float* __restrict__ q_bias,
                       const float* __restrict__ v_bias,
                       const float* __restrict__ logit_scale,
                       const float* __restrict__ proj_w,
                       const float* __restrict__ proj_b,
                       const float* __restrict__ bias_table,
                       const int*   __restrict__ rel_idx,
                       float* __restrict__ out) {
    extern __shared__ float sm[];
    float* xs   = sm;                    // [64][XS]      x tile
    float* qs   = xs  + 64 * XS;         // [2][64][QS]   q for 2 heads
    float* ks   = qs  + 2 * 64 * QS;     // [2][64][QS]
    float* vs   = ks  + 2 * 64 * QS;     // [2][64][QS]
    float* at   = vs  + 2 * 64 * QS;     // [2][64][AS]   attention probs
    float* outs = at  + 2 * 64 * AS;     // [64][XS]      concat head outputs

    const int b    = blockIdx.x;
    const int tid  = threadIdx.x;
    const int wave = tid >> 5;
    const int lane = tid & 31;
    const int lrow = lane & 15;   // M/N index within a 16-wide fragment
    const int lhi  = lane >> 4;   // K half-select (K = 4: lanes 0-15 -> k0,k1)

    // ---- stage x[b] (49x384) into LDS, zero the v pad rows (K-padding) ----
    const float* xb = x + (size_t)b * NTOK * DIMC;
    for (int i = tid; i < (NTOK * DIMC) / 4; i += 384) {
        int e = i * 4;
        int r = e / DIMC, c = e % DIMC;
        const float4 v = *(const float4*)(xb + e);
        float* d = xs + r * XS + c;
        d[0] = v.x; d[1] = v.y; d[2] = v.z; d[3] = v.w;
    }
    for (int i = tid; i < 2 * 15 * QS; i += 384) {
        int hl = i / (15 * QS);
        int rr = (i / QS) % 15;
        int cc = i % QS;
        vs[hl * 64 * QS + (NTOK + rr) * QS + cc] = 0.f;   // rows 49..63 := 0
    }
    __syncthreads();

    const float* mrow = mask + (size_t)(b % NWIN) * NTOK * NTOK;

    // ================= head pairs: 6 x (2 heads) =================
    for (int p = 0; p < 6; ++p) {
        const int h0 = 2 * p;

        // ---- QKV GEMM: 12 jobs = 2 heads x {q,k,v} x 2 N-tiles ----
        {
            int hl   = wave / 6;        // which head of the pair
            int rem  = wave % 6;
            int mat  = rem >> 1;        // 0=q 1=k 2=v
            int N0   = rem & 1;
            int head = h0 + hl;
            int row0 = mat * DIMC + head * HD;
            int n    = N0 * 16 + lrow;  // 0..31 output feature
            const float* wrow = qkv_w + (size_t)(row0 + n) * DIMC;

            v8f acc0 = {0,0,0,0,0,0,0,0};
            v8f acc1 = acc0, acc2 = acc0, acc3 = acc0;
            for (int kk = 0; kk < 96; ++kk) {
                int k0 = kk * 4 + 2 * lhi;
                v2f bf; bf.x = wrow[k0]; bf.y = wrow[k0 + 1];
                v2f a0; a0.x = xs[(0 * 16 + lrow) * XS + k0]; a0.y = xs[(0 * 16 + lrow) * XS + k0 + 1];
                v2f a1; a1.x = xs[(1 * 16 + lrow) * XS + k0]; a1.y = xs[(1 * 16 + lrow) * XS + k0 + 1];
                v2f a2; a2.x = xs[(2 * 16 + lrow) * XS + k0]; a2.y = xs[(2 * 16 + lrow) * XS + k0 + 1];
                v2f a3; a3.x = xs[(3 * 16 + lrow) * XS + k0]; a3.y = xs[(3 * 16 + lrow) * XS + k0 + 1];
                acc0 = wmma4(a0, bf, acc0);
                acc1 = wmma4(a1, bf, acc1);
                acc2 = wmma4(a2, bf, acc2);
                acc3 = wmma4(a3, bf, acc3);
            }
            float bias = (mat == 0) ? q_bias[head * HD + n]
                       : (mat == 2) ? v_bias[head * HD + n] : 0.f;
            float* dst = (mat == 0 ? qs : (mat == 1 ? ks : vs)) + hl * 64 * QS;
            v8f accs[4] = {acc0, acc1, acc2, acc3};
            #pragma unroll
            for (int M0 = 0; M0 < 4; ++M0)
                #pragma unroll
                for (int r = 0; r < 8; ++r) {
                    int m = M0 * 16 + r + 8 * lhi;       // documented C/D layout
                    if (m < NTOK) dst[m * QS + n] = accs[M0][r] + bias;
                }
        }
        __syncthreads();

        // ---- cosine-attn normalization + logit scale (196 row jobs) ----
        if (tid < 196) {
            int which = tid / 98;             // 0=q rows, 1=k rows
            int rr    = tid % 98;
            int hl    = rr / NTOK, row = rr % NTOK;
            float* base = (which == 0 ? qs : ks) + hl * 64 * QS + row * QS;
            float ss = 0.f;
            #pragma unroll
            for (int d2 = 0; d2 < HD; ++d2) { float t2 = base[d2]; ss += t2 * t2; }
            float inv = 1.0f / fmaxf(sqrtf(ss), 1e-12f);
            if (which == 0) {
                float ls = logit_scale[h0 + hl];
                inv *= expf(fminf(ls, 4.6051701859880914f));   // ln(100)
            }
            #pragma unroll
            for (int d2 = 0; d2 < HD; ++d2) base[d2] *= inv;
        }
        __syncthreads();

        // ---- attn = qn @ kn^T  (2 heads x 16 tiles = 32 jobs) ----
        for (int job = wave; job < 32; job += 12) {
            int hl = job >> 4;
            int t  = job & 15;
            int M0 = t >> 2, N0 = t & 3;
            const float* qb = qs + hl * 64 * QS;
            const float* kb = ks + hl * 64 * QS;
            int mA = M0 * 16 + lrow;
            int nB = N0 * 16 + lrow;
            v8f acc = {0,0,0,0,0,0,0,0};
            #pragma unroll
            for (int kk = 0; kk < 8; ++kk) {               // K = 32
                int k0 = kk * 4 + 2 * lhi;
                v2f a;  a.x  = qb[mA * QS + k0]; a.y  = qb[mA * QS + k0 + 1];
                v2f bb; bb.x = kb[nB * QS + k0]; bb.y = kb[nB * QS + k0 + 1];
                acc = wmma4(a, bb, acc);
            }
            int h = h0 + hl;
            int n = N0 * 16 + lrow;
            #pragma unroll
            for (int r = 0; r < 8; ++r) {
                int m = M0 * 16 + r + 8 * lhi;
                if (m < NTOK && n < NTOK) {
                    int   ri   = rel_idx[m * NTOK + n];
                    float bias = bias_table[ri * NH + h] + mrow[m * NTOK + n];
                    at[hl * 64 * AS + m * AS + n] = acc[r] + bias;
                }
            }
        }
        __syncthreads();

        // ---- softmax over keys (98 row jobs); zero K-pad cols 49..51 ----
        if (tid < 98) {
            int hl = tid / NTOK, row = tid % NTOK;
            float* arow = at + hl * 64 * AS + row * AS;
            float mx = -1e30f;
            for (int j = 0; j < NTOK; ++j) mx = fmaxf(mx, arow[j]);
            float s = 0.f;
            for (int j = 0; j < NTOK; ++j) { float e = expf(arow[j] - mx); arow[j] = e; s += e; }
            float inv = 1.f / s;
            for (int j = 0; j < NTOK; ++j) arow[j] *= inv;
            arow[49] = 0.f; arow[50] = 0.f; arow[51] = 0.f;
        }
        __syncthreads();

        // ---- out_h = attn @ v  (2 heads x 8 tiles = 16 jobs, K padded to 52) ----
        for (int job = wave; job < 16; job += 12) {
            int hl = job >> 3;
            int t  = job & 7;
            int M0 = t >> 1, N0 = t & 1;
            const float* ab = at + hl * 64 * AS;
            const float* vb = vs + hl * 64 * QS;
            int mA = M0 * 16 + lrow;
            int n  = N0 * 16 + lrow;
            v8f acc = {0,0,0,0,0,0,0,0};
            #pragma unroll
            for (int kk = 0; kk < 13; ++kk) {              // K = 52 (padded)
                int k0 = kk * 4 + 2 * lhi;
                v2f a;  a.x  = ab[mA * AS + k0];  a.y  = ab[mA * AS + k0 + 1];
                v2f bb; bb.x = vb[k0 * QS + n];   bb.y = vb[(k0 + 1) * QS + n];
                acc = wmma4(a, bb, acc);
            }
            int h = h0 + hl;
            #pragma unroll
            for (int r = 0; r < 8; ++r) {
                int m = M0 * 16 + r + 8 * lhi;
                if (m < NTOK) outs[m * XS + h * HD + n] = acc[r];
            }
        }
        __syncthreads();
    }

    // ---- final projection: out = outs @ proj_w^T + proj_b ----
    // 24 N-tiles x 4 M-tiles; each wave owns 2 N-tiles (B fragment reused
    // across the 4 M accumulators -> each weight element hits L2 once/block).
    float* ob = out + (size_t)b * NTOK * DIMC;
    for (int nn = wave * 2; nn < wave * 2 + 2; ++nn) {
        int n = nn * 16 + lrow;                            // 0..383
        const float* wrow = proj_w + (size_t)n * DIMC;
        v8f acc0 = {0,0,0,0,0,0,0,0};
        v8f acc1 = acc0, acc2 = acc0, acc3 = acc0;
        for (int kk = 0; kk < 96; ++kk) {
            int k0 = kk * 4 + 2 * lhi;
            v2f bf; bf.x = wrow[k0]; bf.y = wrow[k0 + 1];
            v2f a0; a0.x = outs[(0 * 16 + lrow) * XS + k0]; a0.y = outs[(0 * 16 + lrow) * XS + k0 + 1];
            v2f a1; a1.x = outs[(1 * 16 + lrow) * XS + k0]; a1.y = outs[(1 * 16 + lrow) * XS + k0 + 1];
            v2f a2; a2.x = outs[(2 * 16 + lrow) * XS + k0]; a2.y = outs[(2 * 16 + lrow) * XS + k0 + 1];
            v2f a3; a3.x = outs[(3 * 16 + lrow) * XS + k0]; a3.y = outs[(3 * 16 + lrow) * XS + k0 + 1];
            acc0 = wmma4(a0, bf, acc0);
            acc1 = wmma4(a1, bf, acc1);
            acc2 = wmma4(a2, bf, acc2);
            acc3 = wmma4(a3, bf, acc3);
        }
        float pb = proj_b[n];
        v8f accs[4] = {acc0, acc1, acc2, acc3};
        #pragma unroll
        for (int M0 = 0; M0 < 4; ++M0)
            #pragma unroll
            for (int r = 0; r < 8; ++r) {
                int m = M0 * 16 + r + 8 * lhi;
                if (m < NTOK) ob[m * DIMC + n] = accs[M0][r] + pb;
            }
    }
}

// ---------------------------------------------------------------------------
extern "C" void kernel_launch(void* const* d_in, const int* in_sizes, int n_in,
                              void* d_out, int out_size, void* d_ws, size_t ws_size,
                              hipStream_t stream) {
    const float* x        = (const float*)d_in[0];
    const float* mask     = (const float*)d_in[1];
    const float* qkv_w    = (const float*)d_in[2];
    const float* q_bias   = (const float*)d_in[3];
    const float* v_bias   = (const float*)d_in[4];
    const float* lscale   = (const float*)d_in[5];
    const float* cpb_w1   = (const float*)d_in[6];
    const float* cpb_b1   = (const float*)d_in[7];
    const float* cpb_w2   = (const float*)d_in[8];
    const float* proj_w   = (const float*)d_in[9];
    const float* proj_b   = (const float*)d_in[10];
    const float* rel_tbl  = (const float*)d_in[11];
    const int*   rel_idx  = (const int*)d_in[12];
    float*       out      = (float*)d_out;

    float* bias_table = (float*)d_ws;   // 169*12 floats = 8112 B

    cpb_table_kernel<<<(TROWS * NH + 255) / 256, 256, 0, stream>>>(
        rel_tbl, cpb_w1, cpb_b1, cpb_w2, bias_table);

    // LDS: xs(64*388) + 3*qkv(2*64*36) + attn(2*64*52) + outs(64*388)
    size_t smem_floats = 64 * XS + 3 * (2 * 64 * QS) + 2 * 64 * AS + 64 * XS;
    size_t smem_bytes  = smem_floats * sizeof(float);   // 280,576 B < 320 KB

    swinv2_attn_fused<<<BWIN, 384, smem_bytes, stream>>>(
        x, mask, qkv_w, q_bias, v_bias, lscale, proj_w, proj_b,
        bias_table, rel_idx, out);
}